// shiftmlp_13443247636656
// MI455X (gfx1250) — compile-verified
//
#include <hip/hip_runtime.h>
#include <math.h>

typedef __attribute__((ext_vector_type(2))) float v2f;
typedef __attribute__((ext_vector_type(8))) float v8f;

#define Bn   8
#define Cn   160
#define Nn   4096
#define HIDn 640
#define BCn  (Bn*Cn)     // 1280
#define BNn  (Bn*Nn)     // 32768

#define LDSP 20          // LDS row pitch (floats): 20 mod 64 -> conflict-free frag reads

// ======================================================================
// Generic 64x64-tile WMMA f32 GEMM:  C = alpha*A*B (+C if accum) (+bias) act
// A: [M,K] row-major (lda).  B: [K,N] row-major (ldb) or, if bTrans,
// [N,K] row-major (ldb=K) i.e. weights stored (out,in).
// Batched via blockIdx.z with element strides sA/sB/sC (0 = shared).
// Requires K % 16 == 0 and lda/ldb % 4 == 0 (true for all call sites).
// act: 0=none 1=relu 2=sigmoid
// LDS tiles are fragment-major ([m][k] / [n][k]) so each WMMA fragment is a
// single aligned ds_load_b64 straight into an even VGPR pair (no repacking).
// ======================================================================
__global__ __launch_bounds__(256)
void gemm64_wmma(const float* __restrict__ A, const float* __restrict__ B,
                 const float* __restrict__ bias, float* __restrict__ C,
                 int M, int N, int K, int lda, int ldb, int ldc,
                 long long sA, long long sB, long long sC,
                 float alpha, int accum, int bTrans, int act)
{
    __shared__ float As[64][LDSP];   // [m][k]
    __shared__ float Bs[64][LDSP];   // [n][k]

    const int tid  = threadIdx.x;
    const int lane = tid & 31;
    const int wave = tid >> 5;
    const int wm   = (wave & 3) << 4;   // wave's row offset in tile (0..48)
    const int wn   = (wave >> 2) << 5;  // wave's col offset in tile (0 or 32)
    const int m0   = blockIdx.y << 6;
    const int n0   = blockIdx.x << 6;

    A += (long long)blockIdx.z * sA;
    B += (long long)blockIdx.z * sB;
    C += (long long)blockIdx.z * sC;

    v8f c0 = {};
    v8f c1 = {};

    const int lm = tid >> 2;          // 0..63
    const int lk = (tid & 3) << 2;    // 0,4,8,12
    const int bn = tid & 63;          // 0..63 (non-trans B)
    const int bq = (tid >> 6) << 2;   // 0,4,8,12 (non-trans B)

    const int r   = lane & 15;
    const int ksl = (lane >> 4) << 1;  // lanes 0-15 -> K+0/1, lanes 16-31 -> K+2/3

    for (int k0 = 0; k0 < K; k0 += 16) {
        // ---- stage A panel into LDS (As[m][k], one b128 store) ----
        {
            float4 v = make_float4(0.f, 0.f, 0.f, 0.f);
            int gm = m0 + lm;
            if (gm < M) {
                v = *(const float4*)(A + (long long)gm * lda + k0 + lk);
                if (k0 + 16 < K)   // hint next K panel toward the caches
                    __builtin_prefetch(A + (long long)gm * lda + k0 + 16 + lk, 0, 3);
            }
            *(float4*)&As[lm][lk] = v;
        }
        // ---- stage B panel into LDS (Bs[n][k], one b128 store) ----
        if (bTrans) {
            float4 v = make_float4(0.f, 0.f, 0.f, 0.f);
            int gn = n0 + lm;
            if (gn < N)
                v = *(const float4*)(B + (long long)gn * ldb + k0 + lk);
            *(float4*)&Bs[lm][lk] = v;
        } else {
            // B row-major [K,N]: 4 n-coalesced b32 loads (k-strided per thread)
            float4 v = make_float4(0.f, 0.f, 0.f, 0.f);
            int gn = n0 + bn;
            if (gn < N) {
                const float* bcol = B + (long long)(k0 + bq) * ldb + gn;
                v.x = bcol[0];
                v.y = bcol[(long long)ldb];
                v.z = bcol[(long long)ldb * 2];
                v.w = bcol[(long long)ldb * 3];
            }
            *(float4*)&Bs[bn][bq] = v;
        }
        __syncthreads();

        // ---- 4 WMMA k-steps of 4: fragments are direct ds_load_b64 ----
#pragma unroll
        for (int kk = 0; kk < 16; kk += 4) {
            const int kb = kk + ksl;
            v2f a  = *(const v2f*)&As[wm + r][kb];
            v2f b0 = *(const v2f*)&Bs[wn + r][kb];
            v2f b1 = *(const v2f*)&Bs[wn + 16 + r][kb];
            c0 = __builtin_amdgcn_wmma_f32_16x16x4_f32(false, a, false, b0,
                                                       (short)0, c0, false, false);
            c1 = __builtin_amdgcn_wmma_f32_16x16x4_f32(false, a, false, b1,
                                                       (short)0, c1, false, false);
        }
        __syncthreads();
    }

    // ---- epilogue: alpha, accumulate, bias, activation ----
    const int col0 = n0 + wn + (lane & 15);
    const int col1 = col0 + 16;
    const int rofs = (lane >> 4) << 3;     // lanes 16-31 hold rows M=r+8
    float bias0 = (bias && col0 < N) ? bias[col0] : 0.f;
    float bias1 = (bias && col1 < N) ? bias[col1] : 0.f;

#pragma unroll
    for (int rr = 0; rr < 8; ++rr) {
        int m = m0 + wm + rofs + rr;
        if (m >= M) continue;
        float* crow = C + (long long)m * ldc;
        if (col0 < N) {
            float v = alpha * c0[rr] + bias0;
            if (accum) v += crow[col0];
            if (act == 1) v = fmaxf(v, 0.f);
            else if (act == 2) v = 1.f / (1.f + expf(-v));
            crow[col0] = v;
        }
        if (col1 < N) {
            float v = alpha * c1[rr] + bias1;
            if (accum) v += crow[col1];
            if (act == 1) v = fmaxf(v, 0.f);
            else if (act == 2) v = 1.f / (1.f + expf(-v));
            crow[col1] = v;
        }
    }
}

// ======================================================================
// DFT matrix init: F[j,k] = exp(-2*pi*i*j*k/64)  ->  Fr=cos, Fi=-sin
// ======================================================================
__global__ void k_init_F(float* __restrict__ Fr, float* __restrict__ Fi)
{
    int i = blockIdx.x * blockDim.x + threadIdx.x;
    if (i >= 4096) return;
    int j = i >> 6, k = i & 63;
    float th = (6.2831853071795864769f / 64.0f) * (float)(j * k);
    Fr[i] = cosf(th);
    Fi[i] = -sinf(th);
}

// Per-(b,c): mean/std (ddof=1), write normalized * (-1)^(h+w) (folds fftshift)
__global__ __launch_bounds__(256)
void k_meanstd_norm(const float* __restrict__ x, float* __restrict__ meanb,
                    float* __restrict__ stdb, float* __restrict__ yp)
{
    __shared__ float s1[256], s2[256];
    int bc = blockIdx.x;
    int b = bc / Cn, c = bc % Cn;
    const float* xb = x + (size_t)b * Nn * Cn + c;
    float sum = 0.f, sq = 0.f;
    for (int n = threadIdx.x; n < Nn; n += 256) {
        float v = xb[(size_t)n * Cn];
        sum += v; sq += v * v;
    }
    s1[threadIdx.x] = sum; s2[threadIdx.x] = sq;
    __syncthreads();
    for (int st = 128; st > 0; st >>= 1) {
        if (threadIdx.x < st) {
            s1[threadIdx.x] += s1[threadIdx.x + st];
            s2[threadIdx.x] += s2[threadIdx.x + st];
        }
        __syncthreads();
    }
    float mean = s1[0] * (1.0f / Nn);
    float var  = (s2[0] - (float)Nn * mean * mean) * (1.0f / (Nn - 1));
    float sd   = sqrtf(fmaxf(var, 0.f));
    if (threadIdx.x == 0) { meanb[bc] = mean; stdb[bc] = sd; }
    float inv = 1.0f / (sd + 1e-6f);
    float* yrow = yp + (size_t)bc * Nn;
    for (int n = threadIdx.x; n < Nn; n += 256) {
        int h = n >> 6, w = n & 63;
        float sgn = ((h + w) & 1) ? -1.f : 1.f;
        yrow[n] = (xb[(size_t)n * Cn] - mean) * inv * sgn;
    }
}

// Frequency masks + noise threshold on high band
__global__ void k_mask(const float* __restrict__ Xre, const float* __restrict__ Xim,
                       const float* __restrict__ alpha, const float* __restrict__ beta,
                       float* __restrict__ Lre, float* __restrict__ Lim,
                       float* __restrict__ Hre, float* __restrict__ Him, int total)
{
    int i = blockIdx.x * blockDim.x + threadIdx.x;
    if (i >= total) return;
    int u = (i >> 6) & 63, v = i & 63;
    float hu = -0.5f + (float)u * (1.0f / 63.0f);
    float hv = -0.5f + (float)v * (1.0f / 63.0f);
    float r  = sqrtf(hu * hu + hv * hv);
    float mid = (r >= 0.2f && r < 0.6f) ? 1.f : 0.f;
    float lm  = (r < 0.2f ? 1.f : 0.f) + alpha[0] * mid;
    float hm  = (r >= 0.6f ? 1.f : 0.f) + beta[0] * mid;
    float xr = Xre[i], xi = Xim[i];
    Lre[i] = xr * lm; Lim[i] = xi * lm;
    float hr = xr * hm, hi = xi * hm;
    if (sqrtf(hr * hr + hi * hi) < 0.1f) { hr = 0.f; hi = 0.f; }
    Hre[i] = hr; Him[i] = hi;
}

// low/high = Re(IFFT)*(-1)^(h+w)/4096 * std + mean
__global__ void k_inv_epilogue(const float* __restrict__ Gl, const float* __restrict__ Gh,
                               const float* __restrict__ meanb, const float* __restrict__ stdb,
                               float* __restrict__ low, float* __restrict__ high, int total)
{
    int i = blockIdx.x * blockDim.x + threadIdx.x;
    if (i >= total) return;
    int bc = i >> 12;
    int h = (i >> 6) & 63, w = i & 63;
    float sc = (((h + w) & 1) ? -1.f : 1.f) * (1.0f / 4096.0f);
    float sd = stdb[bc], mn = meanb[bc];
    low[i]  = Gl[i] * sc * sd + mn;
    high[i] = Gh[i] * sc * sd + mn;
}

// zero-padded diagonal per-chunk shift of low -> feat[:, 0:160]
__global__ void k_lowshift(const float* __restrict__ low, const int* __restrict__ shifts,
                           float* __restrict__ feat, int total)
{
    int i = blockIdx.x * blockDim.x + threadIdx.x;
    if (i >= total) return;
    int n = i & 4095;
    int c = (i >> 12) % Cn;
    int b = (i >> 12) / Cn;
    int s = shifts[c >> 5];             // chunk size 160/5 = 32
    int h = n >> 6, w = n & 63;
    int hh = h - s, ww = w - s;
    float v = 0.f;
    if (hh >= 0 && hh < 64 && ww >= 0 && ww < 64)
        v = low[((size_t)(b * Cn + c) << 12) + (hh << 6) + ww];
    feat[((size_t)(b << 12) + n) * 320 + c] = v;
}

// edge = mean over channels of high
__global__ void k_edge(const float* __restrict__ high, float* __restrict__ edge)
{
    int i = blockIdx.x * blockDim.x + threadIdx.x;
    if (i >= BNn) return;
    int b = i >> 12, n = i & 4095;
    float s = 0.f;
    const float* base = high + ((size_t)b * Cn << 12) + n;
    for (int c = 0; c < Cn; ++c) s += base[(size_t)c << 12];
    edge[i] = s * (1.0f / Cn);
}

// conv 1->16, 3x3 SAME, ReLU
__global__ void k_conv1(const float* __restrict__ edge, const float* __restrict__ cw,
                        const float* __restrict__ cb, float* __restrict__ t16)
{
    int i = blockIdx.x * blockDim.x + threadIdx.x;
    if (i >= Bn * 16 * Nn) return;
    int n = i & 4095;
    int o = (i >> 12) & 15;
    int b = i >> 16;
    int h = n >> 6, w = n & 63;
    const float* e = edge + ((size_t)b << 12);
    float acc = cb[o];
#pragma unroll
    for (int di = -1; di <= 1; ++di)
#pragma unroll
        for (int dj = -1; dj <= 1; ++dj) {
            int hh = h + di, ww = w + dj;
            if (hh >= 0 && hh < 64 && ww >= 0 && ww < 64)
                acc += cw[o * 9 + (di + 1) * 3 + (dj + 1)] * e[(hh << 6) + ww];
        }
    t16[i] = fmaxf(acc, 0.f);
}

// conv 16->2, 3x3 SAME, fused 2-way softmax -> dir0 (dir1 = 1-dir0)
__global__ void k_conv2(const float* __restrict__ t16, const float* __restrict__ cw,
                        const float* __restrict__ cb, float* __restrict__ dir0)
{
    int i = blockIdx.x * blockDim.x + threadIdx.x;
    if (i >= BNn) return;
    int b = i >> 12, n = i & 4095;
    int h = n >> 6, w = n & 63;
    float d[2] = { cb[0], cb[1] };
    for (int o = 0; o < 16; ++o) {
        const float* tin = t16 + (((size_t)b * 16 + o) << 12);
#pragma unroll
        for (int di = -1; di <= 1; ++di)
#pragma unroll
            for (int dj = -1; dj <= 1; ++dj) {
                int hh = h + di, ww = w + dj;
                if (hh >= 0 && hh < 64 && ww >= 0 && ww < 64) {
                    float v = tin[(hh << 6) + ww];
                    int tap = (di + 1) * 3 + (dj + 1);
                    d[0] += cw[(0 * 16 + o) * 9 + tap] * v;
                    d[1] += cw[(1 * 16 + o) * 9 + tap] * v;
                }
            }
    }
    dir0[i] = 1.f / (1.f + expf(d[1] - d[0]));   // softmax over 2 channels
}

// per-chunk H-shift and W-shift of high, blended by dirw -> feat[:, 160:320]
__global__ void k_highshift(const float* __restrict__ high, const int* __restrict__ sh_h,
                            const int* __restrict__ sh_w, const float* __restrict__ dir0,
                            float* __restrict__ feat, int total)
{
    int i = blockIdx.x * blockDim.x + threadIdx.x;
    if (i >= total) return;
    int n = i & 4095;
    int c = (i >> 12) % Cn;
    int b = (i >> 12) / Cn;
    int h = n >> 6, w = n & 63;
    const float* base = high + ((size_t)(b * Cn + c) << 12);
    int s1 = sh_h[c >> 5], s2 = sh_w[c >> 5];
    int hh = h - s1, ww = w - s2;
    float vh = (hh >= 0 && hh < 64) ? base[(hh << 6) + w] : 0.f;
    float vw = (ww >= 0 && ww < 64) ? base[(h << 6) + ww] : 0.f;
    float d0 = dir0[(b << 12) + n];
    feat[((size_t)(b << 12) + n) * 320 + 160 + c] = d0 * vh + (1.f - d0) * vw;
}

// fus2 linear + 3-way softmax + fused = w0*xl + w1*xh + w2*(xl*xh)
__global__ void k_fuse(const float* __restrict__ w1, const float* __restrict__ fw,
                       const float* __restrict__ fb, const float* __restrict__ feat,
                       float* __restrict__ fused)
{
    int i = blockIdx.x * blockDim.x + threadIdx.x;
    if (i >= BNn) return;
    const float* wr = w1 + (size_t)i * 40;
    float l0 = fb[0], l1 = fb[1], l2 = fb[2];
    for (int j = 0; j < 40; ++j) {
        float v = wr[j];
        l0 += fw[j] * v; l1 += fw[40 + j] * v; l2 += fw[80 + j] * v;
    }
    float mx = fmaxf(l0, fmaxf(l1, l2));
    float e0 = expf(l0 - mx), e1 = expf(l1 - mx), e2 = expf(l2 - mx);
    float inv = 1.f / (e0 + e1 + e2);
    e0 *= inv; e1 *= inv; e2 *= inv;
    const float* fr = feat + (size_t)i * 320;
    float* out = fused + (size_t)i * Cn;
    for (int c = 0; c < Cn; ++c) {
        float xl = fr[c], xh = fr[160 + c];
        out[c] = e0 * xl + e1 * xh + e2 * xl * xh;
    }
}

// x2 = x*g + fused*(1-g)
__global__ void k_gatecombine(const float* __restrict__ x, const float* __restrict__ g,
                              const float* __restrict__ fused, float* __restrict__ x2, int total)
{
    int i = blockIdx.x * blockDim.x + threadIdx.x;
    if (i >= total) return;
    float gv = g[i];
    x2[i] = x[i] * gv + fused[i] * (1.f - gv);
}

// depthwise 3x3 SAME over [B,HID,H,W] view of h1 ([B,N,HID] layout) + exact GELU
__global__ void k_dwgelu(const float* __restrict__ h1, const float* __restrict__ dw,
                         const float* __restrict__ db, float* __restrict__ h2, int total)
{
    int i = blockIdx.x * blockDim.x + threadIdx.x;
    if (i >= total) return;
    int o = i % HIDn;
    int r = i / HIDn;
    int n = r & 4095, b = r >> 12;
    int h = n >> 6, w = n & 63;
    float acc = db[o];
#pragma unroll
    for (int di = -1; di <= 1; ++di)
#pragma unroll
        for (int dj = -1; dj <= 1; ++dj) {
            int hh = h + di, ww = w + dj;
            if (hh >= 0 && hh < 64 && ww >= 0 && ww < 64)
                acc += dw[o * 9 + (di + 1) * 3 + (dj + 1)] *
                       h1[(((size_t)b << 12) + (hh << 6) + ww) * HIDn + o];
        }
    h2[i] = 0.5f * acc * (1.0f + erff(acc * 0.70710678118654752f));
}

// fixed diagonal per-chunk shift over HID channels (chunk 128, shift = chunk-2)
__global__ void k_diagshift(const float* __restrict__ h2, float* __restrict__ xs, int total)
{
    int i = blockIdx.x * blockDim.x + threadIdx.x;
    if (i >= total) return;
    int o = i % HIDn;
    int r = i / HIDn;
    int n = r & 4095, b = r >> 12;
    int s = (o >> 7) - 2;
    int h = n >> 6, w = n & 63;
    int hh = h - s, ww = w - s;
    float v = 0.f;
    if (hh >= 0 && hh < 64 && ww >= 0 && ww < 64)
        v = h2[(((size_t)b << 12) + (hh << 6) + ww) * HIDn + o];
    xs[i] = v;
}

// ======================================================================
// Host orchestration
// ======================================================================
static inline void gemm_go(hipStream_t s, const float* A, const float* B, const float* bias,
                           float* C, int M, int N, int K, int lda, int ldb, int ldc,
                           long long sA, long long sB, long long sC, int batch,
                           float alpha, int accum, int bTrans, int act)
{
    dim3 grid((N + 63) / 64, (M + 63) / 64, batch);
    gemm64_wmma<<<grid, 256, 0, s>>>(A, B, bias, C, M, N, K, lda, ldb, ldc,
                                     sA, sB, sC, alpha, accum, bTrans, act);
}

extern "C" void kernel_launch(void* const* d_in, const int* in_sizes, int n_in,
                              void* d_out, int out_size, void* d_ws, size_t ws_size,
                              hipStream_t stream)
{
    const float* x      = (const float*)d_in[0];
    const float* alpha  = (const float*)d_in[1];
    const float* beta   = (const float*)d_in[2];
    const float* hc1_w  = (const float*)d_in[3];
    const float* hc1_b  = (const float*)d_in[4];
    const float* hc2_w  = (const float*)d_in[5];
    const float* hc2_b  = (const float*)d_in[6];
    const float* fus1_w = (const float*)d_in[7];
    const float* fus1_b = (const float*)d_in[8];
    const float* fus2_w = (const float*)d_in[9];
    const float* fus2_b = (const float*)d_in[10];
    const float* gate_w = (const float*)d_in[11];
    const float* gate_b = (const float*)d_in[12];
    const float* fc1_w  = (const float*)d_in[13];
    const float* fc1_b  = (const float*)d_in[14];
    const float* dw_w   = (const float*)d_in[15];
    const float* dw_b   = (const float*)d_in[16];
    const float* fc2_w  = (const float*)d_in[17];
    const float* fc2_b  = (const float*)d_in[18];
    // d_in[19]=H, d_in[20]=W (compile-time constants here)
    const int* low_shifts = (const int*)d_in[21];
    const int* sh_h       = (const int*)d_in[22];
    const int* sh_w       = (const int*)d_in[23];

    // ---- workspace carving (floats), 64-element aligned ----
    float* wsp = (float*)d_ws;
    size_t off = 0;
    auto take = [&](size_t n) { float* p = wsp + off; off += (n + 63) & ~(size_t)63; return p; };
    const size_t IMG = (size_t)BCn * Nn;   // 5,242,880
    float* Fr    = take(4096);
    float* Fi    = take(4096);
    float* meanb = take(BCn);
    float* stdb  = take(BCn);
    float* P0 = take(IMG); float* P1 = take(IMG); float* P2 = take(IMG);
    float* P3 = take(IMG); float* P4 = take(IMG); float* P5 = take(IMG);
    float* P6 = take(IMG); float* P7 = take(IMG); float* P8 = take(IMG);
    float* feat = take((size_t)BNn * 320);
    float* edge = take(BNn);
    float* t16  = take((size_t)Bn * 16 * Nn);
    float* dir0 = take(BNn);
    float* w1b  = take((size_t)BNn * 40);
    float* h1b  = take((size_t)BNn * HIDn);
    float* h2b  = take((size_t)BNn * HIDn);
    (void)ws_size; (void)in_sizes; (void)n_in; (void)out_size;

    // aliases (sequenced reuse)
    float* Yp = P0, *Rre = P1, *Rim = P2, *Xre = P3, *Xim = P4;
    float* Lre = P5, *Lim = P6, *Hre = P7, *Him = P8;
    float* Ure = P0, *Uim = P1, *Glow = P2, *Ghigh = P3;
    float* lowb = P4, *highb = P6;
    float* fusedb = P7, *gbuf = P8, *x2b = P0;
    float* xsb = h1b;   // diag-shift output reuses h1

    dim3 blk(256);
    auto g1 = [](int total) { return dim3((total + 255) / 256); };

    // ---- DFT matrices + normalization (fftshift folded as (-1)^(h+w)) ----
    k_init_F<<<g1(4096), blk, 0, stream>>>(Fr, Fi);
    k_meanstd_norm<<<dim3(BCn), blk, 0, stream>>>(x, meanb, stdb, Yp);

    // ---- forward 2D DFT as batched GEMMs: X = F * (Yp * F) ----
    gemm_go(stream, Yp, Fr, nullptr, Rre, 64, 64, 64, 64, 64, 64, Nn, 0, Nn, BCn,  1.f, 0, 0, 0);
    gemm_go(stream, Yp, Fi, nullptr, Rim, 64, 64, 64, 64, 64, 64, Nn, 0, Nn, BCn,  1.f, 0, 0, 0);
    gemm_go(stream, Fr, Rre, nullptr, Xre, 64, 64, 64, 64, 64, 64, 0, Nn, Nn, BCn, 1.f, 0, 0, 0);
    gemm_go(stream, Fi, Rim, nullptr, Xre, 64, 64, 64, 64, 64, 64, 0, Nn, Nn, BCn, -1.f, 1, 0, 0);
    gemm_go(stream, Fr, Rim, nullptr, Xim, 64, 64, 64, 64, 64, 64, 0, Nn, Nn, BCn, 1.f, 0, 0, 0);
    gemm_go(stream, Fi, Rre, nullptr, Xim, 64, 64, 64, 64, 64, 64, 0, Nn, Nn, BCn, 1.f, 1, 0, 0);

    // ---- masks + noise threshold ----
    k_mask<<<g1((int)IMG), blk, 0, stream>>>(Xre, Xim, alpha, beta, Lre, Lim, Hre, Him, (int)IMG);

    // ---- inverse DFT (low):  G = F^H * (Z * F^H),  F^H = Fr - i*Fi ----
    gemm_go(stream, Lre, Fr, nullptr, Ure, 64, 64, 64, 64, 64, 64, Nn, 0, Nn, BCn,  1.f, 0, 0, 0);
    gemm_go(stream, Lim, Fi, nullptr, Ure, 64, 64, 64, 64, 64, 64, Nn, 0, Nn, BCn,  1.f, 1, 0, 0);
    gemm_go(stream, Lim, Fr, nullptr, Uim, 64, 64, 64, 64, 64, 64, Nn, 0, Nn, BCn,  1.f, 0, 0, 0);
    gemm_go(stream, Lre, Fi, nullptr, Uim, 64, 64, 64, 64, 64, 64, Nn, 0, Nn, BCn, -1.f, 1, 0, 0);
    gemm_go(stream, Fr, Ure, nullptr, Glow, 64, 64, 64, 64, 64, 64, 0, Nn, Nn, BCn, 1.f, 0, 0, 0);
    gemm_go(stream, Fi, Uim, nullptr, Glow, 64, 64, 64, 64, 64, 64, 0, Nn, Nn, BCn, 1.f, 1, 0, 0);
    // ---- inverse DFT (high) ----
    gemm_go(stream, Hre, Fr, nullptr, Ure, 64, 64, 64, 64, 64, 64, Nn, 0, Nn, BCn,  1.f, 0, 0, 0);
    gemm_go(stream, Him, Fi, nullptr, Ure, 64, 64, 64, 64, 64, 64, Nn, 0, Nn, BCn,  1.f, 1, 0, 0);
    gemm_go(stream, Him, Fr, nullptr, Uim, 64, 64, 64, 64, 64, 64, Nn, 0, Nn, BCn,  1.f, 0, 0, 0);
    gemm_go(stream, Hre, Fi, nullptr, Uim, 64, 64, 64, 64, 64, 64, Nn, 0, Nn, BCn, -1.f, 1, 0, 0);
    gemm_go(stream, Fr, Ure, nullptr, Ghigh, 64, 64, 64, 64, 64, 64, 0, Nn, Nn, BCn, 1.f, 0, 0, 0);
    gemm_go(stream, Fi, Uim, nullptr, Ghigh, 64, 64, 64, 64, 64, 64, 0, Nn, Nn, BCn, 1.f, 1, 0, 0);

    k_inv_epilogue<<<g1((int)IMG), blk, 0, stream>>>(Glow, Ghigh, meanb, stdb, lowb, highb, (int)IMG);

    // ---- shifts + edge guidance ----
    k_lowshift<<<g1((int)IMG), blk, 0, stream>>>(lowb, low_shifts, feat, (int)IMG);
    k_edge<<<g1(BNn), blk, 0, stream>>>(highb, edge);
    k_conv1<<<g1(Bn * 16 * Nn), blk, 0, stream>>>(edge, hc1_w, hc1_b, t16);
    k_conv2<<<g1(BNn), blk, 0, stream>>>(t16, hc2_w, hc2_b, dir0);
    k_highshift<<<g1((int)IMG), blk, 0, stream>>>(highb, sh_h, sh_w, dir0, feat, (int)IMG);

    // ---- fusion: w1 = relu(feat @ fus1_w^T + b) via WMMA, then softmax blend ----
    gemm_go(stream, feat, fus1_w, fus1_b, w1b, BNn, 40, 320, 320, 320, 40, 0, 0, 0, 1, 1.f, 0, 1, 1);
    k_fuse<<<g1(BNn), blk, 0, stream>>>(w1b, fus2_w, fus2_b, feat, fusedb);

    // ---- gate (sigmoid fused into GEMM epilogue) + combine ----
    gemm_go(stream, x, gate_w, gate_b, gbuf, BNn, Cn, Cn, Cn, Cn, Cn, 0, 0, 0, 1, 1.f, 0, 1, 2);
    k_gatecombine<<<g1((int)IMG), blk, 0, stream>>>(x, gbuf, fusedb, x2b, (int)IMG);

    // ---- MLP: fc1 -> depthwise+GELU -> diagonal shift -> fc2 ----
    gemm_go(stream, x2b, fc1_w, fc1_b, h1b, BNn, HIDn, Cn, Cn, Cn, HIDn, 0, 0, 0, 1, 1.f, 0, 1, 0);
    k_dwgelu<<<g1(BNn * HIDn), blk, 0, stream>>>(h1b, dw_w, dw_b, h2b, BNn * HIDn);
    k_diagshift<<<g1(BNn * HIDn), blk, 0, stream>>>(h2b, xsb, BNn * HIDn);
    gemm_go(stream, xsb, fc2_w, fc2_b, (float*)d_out, BNn, Cn, HIDn, HIDn, HIDn, Cn, 0, 0, 0, 1, 1.f, 0, 1, 0);
}